// NeuralSemiLagrangian_79912161509635
// MI455X (gfx1250) — compile-verified
//
#include <hip/hip_runtime.h>
#include <hip/hip_bf16.h>

// Problem constants (B=1)
#define HH   360
#define WW   720
#define CC   256
#define NVV  16
#define HWSZ (HH * WW)          // 259200
#define PADP 2
#define HP   (HH + 2 * PADP)    // 364
#define WP   (WW + 2 * PADP)    // 724
#define EPSF 1e-5f
#define ACUB (-0.75f)

typedef __attribute__((ext_vector_type(2))) float v2f;
typedef __attribute__((ext_vector_type(8))) float v8f;

__device__ __forceinline__ int iclamp(int v, int lo, int hi) {
    return v < lo ? lo : (v > hi ? hi : v);
}

// ---------------------------------------------------------------------------
// Kernel 1: per-pixel LayerNorm statistics (mean, rsqrt(var+eps)) over C=256
// ---------------------------------------------------------------------------
__global__ void k_stats(const float* __restrict__ x,
                        float* __restrict__ mu, float* __restrict__ rsig) {
    int p = blockIdx.x * blockDim.x + threadIdx.x;
    if (p >= HWSZ) return;
    float s = 0.f, s2 = 0.f;
    for (int c = 0; c < CC; ++c) {
        float v = x[(size_t)c * HWSZ + p];
        s += v; s2 += v * v;
    }
    float m   = s * (1.f / CC);
    float var = s2 * (1.f / CC) - m * m;
    mu[p]   = m;
    rsig[p] = rsqrtf(var + EPSF);
}

// ---------------------------------------------------------------------------
// Kernel 2: fused LN + geo_pad(1) + depthwise 3x3 conv -> dw
// geo pad(1): row -1 -> row 0 rolled by W/2; row H -> row H-1 rolled; lon wraps
// ---------------------------------------------------------------------------
__global__ void k_dwconv(const float* __restrict__ x,
                         const float* __restrict__ mu, const float* __restrict__ rsig,
                         const float* __restrict__ ln_g, const float* __restrict__ ln_b,
                         const float* __restrict__ dw_w, const float* __restrict__ dw_b,
                         float* __restrict__ dw) {
    int idx = blockIdx.x * blockDim.x + threadIdx.x;   // c*HW + p (exact grid)
    int c = idx / HWSZ;
    int p = idx - c * HWSZ;
    int y = p / WW, xx = p - y * WW;
    float g = ln_g[c], bb = ln_b[c];
    float acc = dw_b[c];
    const float* wc = dw_w + c * 9;
    const float* xc = x + (size_t)c * HWSZ;
#pragma unroll
    for (int ky = 0; ky < 3; ++ky) {
        int ny = y + ky - 1;
        int row; bool pole;
        if (ny < 0)        { row = 0;      pole = true;  }
        else if (ny >= HH) { row = HH - 1; pole = true;  }
        else               { row = ny;     pole = false; }
#pragma unroll
        for (int kx = 0; kx < 3; ++kx) {
            int nx = xx + kx - 1;
            if (nx < 0) nx += WW; else if (nx >= WW) nx -= WW;
            int col = pole ? (nx >= WW / 2 ? nx - WW / 2 : nx + WW / 2) : nx;
            int q = row * WW + col;
            float v = (xc[q] - mu[q]) * rsig[q] * g + bb;
            acc += v * wc[ky * 3 + kx];
        }
    }
    dw[idx] = acc;
}

// ---------------------------------------------------------------------------
// Kernel 3: WMMA fp32 GEMMs.
//   blockIdx.y = 0,1 : vel rows [0..15],[16..31] = pw_w(32x256) x dw(256xHW)
//   blockIdx.y = 2   : proj rows [0..15]         = w_down(16x256) x  x(256xHW)
// One wave (32 lanes) per 16-pixel N tile; K=256 in steps of 4 via
// V_WMMA_F32_16X16X4_F32.  A fragment: lane holds row M=lane&15, VGPR0=K+2h,
// VGPR1=K+2h+1 (h = lane>>4).  B mirrored with lanes = N columns.
// ---------------------------------------------------------------------------
__global__ void k_gemm3(const float* __restrict__ dw, const float* __restrict__ x,
                        const float* __restrict__ pw_w, const float* __restrict__ pw_b,
                        const float* __restrict__ w_down, const float* __restrict__ b_down,
                        float* __restrict__ vel, float* __restrict__ proj) {
    int lane = threadIdx.x & 31;
    int wv   = threadIdx.x >> 5;
    int tile = blockIdx.x * (blockDim.x >> 5) + wv;   // N tile (exact: 16200)
    int my   = blockIdx.y;
    int n0   = tile * 16;
    int l16  = lane & 15;
    int half = lane >> 4;

    const float* In  = (my == 2) ? x      : dw;
    const float* Wt  = (my == 2) ? w_down : pw_w;
    const float* Bs  = (my == 2) ? b_down : pw_b;
    float*       Out = (my == 2) ? proj   : vel;
    int m0 = (my == 1) ? 16 : 0;

    const float* wrow = Wt + (size_t)(m0 + l16) * CC;
    const float* bcol = In + n0 + l16;

    v8f acc = {};
    for (int k = 0; k < CC; k += 4) {
        int kk = k + 2 * half;
        v2f a = *(const v2f*)(wrow + kk);           // A: 16x4 f32 fragment
        v2f b;
        b.x = bcol[(size_t)kk * HWSZ];              // B: 4x16 f32 fragment
        b.y = bcol[(size_t)(kk + 1) * HWSZ];
        acc = __builtin_amdgcn_wmma_f32_16x16x4_f32(
            false, a, false, b, (short)0, acc, false, false);
    }
#pragma unroll
    for (int r = 0; r < 8; ++r) {
        int m = m0 + r + 8 * half;                  // D: VGPR r, lane half -> M
        Out[(size_t)m * HWSZ + n0 + l16] = acc[r] + Bs[m];
    }
}

// ---------------------------------------------------------------------------
// Kernel 4: semi-Lagrangian departure points -> normalized grid coords gx,gy
// ---------------------------------------------------------------------------
__global__ void k_depart(const float* __restrict__ vel,
                         const float* __restrict__ lat_grid,
                         const float* __restrict__ lon_grid,
                         const float* __restrict__ dtp,
                         float* __restrict__ gx, float* __restrict__ gy) {
    int idx = blockIdx.x * blockDim.x + threadIdx.x;  // nv*HW + p (exact grid)
    int nv = idx / HWSZ;
    int p  = idx - nv * HWSZ;
    float dtv = dtp[0];
    float u = vel[(size_t)nv * HWSZ + p];
    float v = vel[(size_t)(NVV + nv) * HWSZ + p];
    float lat_p = lat_grid[p], lon_p = lon_grid[p];

    float lon_pr = -u * dtv, lat_pr = -v * dtv;
    float s_ltp = sinf(lat_pr), c_ltp = cosf(lat_pr);
    float s_lnp = sinf(lon_pr), c_lnp = cosf(lon_pr);
    float sl = sinf(lat_p), cl = cosf(lat_p);

    float sin_lat = s_ltp * cl + c_ltp * c_lnp * sl;
    sin_lat = fminf(fmaxf(sin_lat, -1.f + 1e-7f), 1.f - 1e-7f);
    float lat_dep = asinf(sin_lat);
    float lon_dep = lon_p + atan2f(c_ltp * s_lnp, c_ltp * c_lnp * cl - s_ltp * sl);
    const float TWO_PI = 6.283185307179586f;
    lon_dep = fmodf(lon_dep + TWO_PI, TWO_PI);      // argument > 0 here

    const float min_lat = -1.5707963267948966f + 0.001f;
    const float max_lat =  1.5707963267948966f - 0.001f;
    const float max_lon = TWO_PI * (float)(WW - 1) / (float)WW;

    float pix_x = lon_dep / max_lon * (WW - 1.f);
    float pix_y = (lat_dep - min_lat) / (max_lat - min_lat) * (HH - 1.f);
    gx[idx] = 2.f * (pix_x + PADP) / (WP - 1.f) - 1.f;
    gy[idx] = 2.f * (pix_y + PADP) / (HP - 1.f) - 1.f;
}

// ---------------------------------------------------------------------------
// Kernel 5: geo_pad(proj, 2) -> padded [NV, HP, WP]
// top pad rows [0,1] <- rows [1,0] rolled by W/2; bottom rows [H+2,H+3] <-
// rows [H-1,H-2] rolled; longitude wrap of 2 columns each side.
// ---------------------------------------------------------------------------
__global__ void k_pad(const float* __restrict__ proj, float* __restrict__ padded) {
    int idx = blockIdx.x * blockDim.x + threadIdx.x;  // nv*HP*WP + iy*WP + jx (exact)
    int nv = idx / (HP * WP);
    int r  = idx - nv * (HP * WP);
    int iy = r / WP, jx = r - iy * WP;
    int col = jx - PADP;
    if (col < 0) col += WW; else if (col >= WW) col -= WW;
    int row; bool pole;
    if      (iy == 0)      { row = 1;      pole = true;  }
    else if (iy == 1)      { row = 0;      pole = true;  }
    else if (iy == HH + 2) { row = HH - 1; pole = true;  }
    else if (iy == HH + 3) { row = HH - 2; pole = true;  }
    else                   { row = iy - PADP; pole = false; }
    if (pole) col = (col >= WW / 2) ? col - WW / 2 : col + WW / 2;
    padded[idx] = proj[(size_t)nv * HWSZ + row * WW + col];
}

// ---------------------------------------------------------------------------
// Kernel 6: bicubic sampling (a=-0.75, align_corners=True, zeros padding)
// ---------------------------------------------------------------------------
__device__ __forceinline__ void cubic_w(float t, float w[4]) {
    float t2 = t * t, t3 = t2 * t;
    w[0] = ACUB * (t3 - 2.f * t2 + t);
    w[1] = (ACUB + 2.f) * t3 - (ACUB + 3.f) * t2 + 1.f;
    float s = 1.f - t;
    w[2] = (ACUB + 2.f) * s * s * s - (ACUB + 3.f) * s * s + 1.f;
    w[3] = 1.f - w[0] - w[1] - w[2];
}

__global__ void k_bicubic(const float* __restrict__ padded,
                          const float* __restrict__ gx, const float* __restrict__ gy,
                          float* __restrict__ samp) {
    int idx = blockIdx.x * blockDim.x + threadIdx.x;  // nv*HW + p (exact grid)
    int nv = idx / HWSZ;
    float ix = (gx[idx] + 1.f) * 0.5f * (WP - 1);
    float iy = (gy[idx] + 1.f) * 0.5f * (HP - 1);
    float ix0f = floorf(ix), iy0f = floorf(iy);
    float wxv[4], wyv[4];
    cubic_w(ix - ix0f, wxv);
    cubic_w(iy - iy0f, wyv);
    int ix0 = (int)ix0f, iy0 = (int)iy0f;
    const float* img = padded + (size_t)nv * (HP * WP);
    float out = 0.f;
#pragma unroll
    for (int dy = 0; dy < 4; ++dy) {
        int yy = iy0 + dy - 1;
        bool okY = (yy >= 0) && (yy < HP);
        int yyc = iclamp(yy, 0, HP - 1);
#pragma unroll
        for (int dx = 0; dx < 4; ++dx) {
            int xx = ix0 + dx - 1;
            bool ok = okY && (xx >= 0) && (xx < WP);
            int xxc = iclamp(xx, 0, WP - 1);
            float v = img[yyc * WP + xxc];
            out += wyv[dy] * wxv[dx] * (ok ? v : 0.f);
        }
    }
    samp[idx] = out;
}

// ---------------------------------------------------------------------------
// Kernel 7: out = w_up(256x16) x samp(16xHW) + b_up  via WMMA fp32, K=16
// ---------------------------------------------------------------------------
__global__ void k_gemm_up(const float* __restrict__ samp,
                          const float* __restrict__ w_up,
                          const float* __restrict__ b_up,
                          float* __restrict__ out) {
    int lane = threadIdx.x & 31;
    int wv   = threadIdx.x >> 5;
    int tile = blockIdx.x * (blockDim.x >> 5) + wv;   // N tile (exact: 16200)
    int m0   = blockIdx.y * 16;                       // 16 M tiles
    int n0   = tile * 16;
    int l16  = lane & 15;
    int half = lane >> 4;

    const float* wrow = w_up + (size_t)(m0 + l16) * NVV;
    const float* bcol = samp + n0 + l16;

    v8f acc = {};
#pragma unroll
    for (int k = 0; k < NVV; k += 4) {
        int kk = k + 2 * half;
        v2f a = *(const v2f*)(wrow + kk);
        v2f b;
        b.x = bcol[(size_t)kk * HWSZ];
        b.y = bcol[(size_t)(kk + 1) * HWSZ];
        acc = __builtin_amdgcn_wmma_f32_16x16x4_f32(
            false, a, false, b, (short)0, acc, false, false);
    }
#pragma unroll
    for (int r = 0; r < 8; ++r) {
        int m = m0 + r + 8 * half;
        out[(size_t)m * HWSZ + n0 + l16] = acc[r] + b_up[m];
    }
}

// ---------------------------------------------------------------------------
// Launch. Workspace layout (bytes), lifetime-aliased; peak ~317 MB:
//   [0)               dw   256*HW*4 = 265,420,800   (live k2 -> k3)
//   [0)               gx    16*HW*4 =  16,588,800   (k4 -> k6, reuses dw)
//   [16,588,800)      gy    16*HW*4                 (k4 -> k6)
//   [33,177,600)      padded 16*HP*WP*4 = 16,866,304 (k5 -> k6)
//   [50,043,904)      samp  16*HW*4                 (k6 -> k7)
//   [265,420,800)     vel   32*HW*4 =  33,177,600   (k3 -> k4)
//   [298,598,400)     proj  16*HW*4                 (k3 -> k5)
//   [315,187,200)     mu     HW*4                   (k1 -> k2)
//   [316,224,000)     rsig   HW*4                   (k1 -> k2)
// ---------------------------------------------------------------------------
extern "C" void kernel_launch(void* const* d_in, const int* in_sizes, int n_in,
                              void* d_out, int out_size, void* d_ws, size_t ws_size,
                              hipStream_t stream) {
    const float* x      = (const float*)d_in[0];
    const float* latg   = (const float*)d_in[1];
    const float* lonG   = (const float*)d_in[2];
    const float* ln_g   = (const float*)d_in[3];
    const float* ln_b   = (const float*)d_in[4];
    const float* dw_w   = (const float*)d_in[5];
    const float* dw_b   = (const float*)d_in[6];
    const float* pw_w   = (const float*)d_in[7];
    const float* pw_b   = (const float*)d_in[8];
    const float* w_down = (const float*)d_in[9];
    const float* b_down = (const float*)d_in[10];
    const float* w_up   = (const float*)d_in[11];
    const float* b_up   = (const float*)d_in[12];
    const float* dtp    = (const float*)d_in[13];

    char* ws = (char*)d_ws;
    float* dw     = (float*)(ws + 0);
    float* gx     = (float*)(ws + 0);
    float* gy     = (float*)(ws + 16588800);
    float* padded = (float*)(ws + 33177600);
    float* samp   = (float*)(ws + 50043904);
    float* vel    = (float*)(ws + 265420800);
    float* proj   = (float*)(ws + 298598400);
    float* mu     = (float*)(ws + 315187200);
    float* rsig   = (float*)(ws + 316224000);

    k_stats <<<dim3((HWSZ + 255) / 256),      dim3(256), 0, stream>>>(x, mu, rsig);
    k_dwconv<<<dim3(CC * HWSZ / 256),         dim3(256), 0, stream>>>(x, mu, rsig, ln_g, ln_b,
                                                                      dw_w, dw_b, dw);
    k_gemm3 <<<dim3(HWSZ / 16 / 8, 3),        dim3(256), 0, stream>>>(dw, x, pw_w, pw_b,
                                                                      w_down, b_down, vel, proj);
    k_depart<<<dim3(NVV * HWSZ / 256),        dim3(256), 0, stream>>>(vel, latg, lonG, dtp, gx, gy);
    k_pad   <<<dim3(NVV * HP * WP / 256),     dim3(256), 0, stream>>>(proj, padded);
    k_bicubic<<<dim3(NVV * HWSZ / 256),       dim3(256), 0, stream>>>(padded, gx, gy, samp);
    k_gemm_up<<<dim3(HWSZ / 16 / 8, CC / 16), dim3(256), 0, stream>>>(samp, w_up, b_up,
                                                                      (float*)d_out);
}